// ImageBasedLight_1047972021055
// MI455X (gfx1250) — compile-verified
//
#include <hip/hip_runtime.h>
#include <stdint.h>

#define MIN_R 0.08f
#define MAX_R 0.5f

typedef float f4 __attribute__((ext_vector_type(4)));

struct F3 { float x, y, z; };
__device__ __forceinline__ F3 operator+(F3 a, F3 b){ return F3{a.x+b.x, a.y+b.y, a.z+b.z}; }
__device__ __forceinline__ F3 operator-(F3 a, F3 b){ return F3{a.x-b.x, a.y-b.y, a.z-b.z}; }
__device__ __forceinline__ F3 operator*(F3 a, F3 b){ return F3{a.x*b.x, a.y*b.y, a.z*b.z}; }
__device__ __forceinline__ F3 operator*(F3 a, float s){ return F3{a.x*s, a.y*s, a.z*s}; }
__device__ __forceinline__ F3 operator+(F3 a, float s){ return F3{a.x+s, a.y+s, a.z+s}; }
__device__ __forceinline__ float dot3(F3 a, F3 b){ return a.x*b.x + a.y*b.y + a.z*b.z; }
__device__ __forceinline__ F3 normalize3(F3 v){
    float r = rsqrtf(fmaxf(dot3(v, v), 1e-20f));
    return v * r;
}
__device__ __forceinline__ int iclamp(int v, int lo, int hi){
    return v < lo ? lo : (v > hi ? hi : v);
}

// Bilinear cube-face sample, identical face/uv math to the JAX reference.
// `cm` may point at global memory (spec mips) or LDS (staged diffuse map);
// address-space inference emits ds_load for the LDS case after inlining.
__device__ __forceinline__ F3 sample_cube(const float* cm, int R, F3 d) {
    float ax = fabsf(d.x), ay = fabsf(d.y), az = fabsf(d.z);
    bool is_x = (ax >= ay) && (ax >= az);
    bool is_y = (!is_x) && (ay >= az);
    int face; float ma, sc, tc;
    if (is_x)      { face = (d.x >= 0.f) ? 0 : 1; ma = ax; sc = (d.x >= 0.f) ? -d.z : d.z; tc = -d.y; }
    else if (is_y) { face = (d.y >= 0.f) ? 2 : 3; ma = ay; sc = d.x;                       tc = (d.y >= 0.f) ? d.z : -d.z; }
    else           { face = (d.z >= 0.f) ? 4 : 5; ma = az; sc = (d.z >= 0.f) ? d.x : -d.x; tc = -d.y; }
    float inv = 1.0f / fmaxf(ma, 1e-12f);
    float fu = (sc * inv + 1.0f) * (0.5f * (float)R) - 0.5f;
    float fv = (tc * inv + 1.0f) * (0.5f * (float)R) - 0.5f;
    float x0f = floorf(fu), y0f = floorf(fv);
    float wx = fu - x0f, wy = fv - y0f;
    int x0 = iclamp((int)x0f, 0, R - 1);
    int y0 = iclamp((int)y0f, 0, R - 1);
    int x1 = iclamp(x0 + 1, 0, R - 1);
    int y1 = iclamp(y0 + 1, 0, R - 1);
    const float* base = cm + (size_t)face * R * R * 3;
    const float* r0 = base + (size_t)y0 * R * 3;
    const float* r1 = base + (size_t)y1 * R * 3;
    F3 c00{r0[3*x0], r0[3*x0+1], r0[3*x0+2]};
    F3 c01{r0[3*x1], r0[3*x1+1], r0[3*x1+2]};
    F3 c10{r1[3*x0], r1[3*x0+1], r1[3*x0+2]};
    F3 c11{r1[3*x1], r1[3*x1+1], r1[3*x1+2]};
    F3 top = c00 * (1.0f - wx) + c01 * wx;
    F3 bot = c10 * (1.0f - wx) + c11 * wx;
    return top * (1.0f - wy) + bot * wy;
}

// Bilinear clamp sample of the 256x256x2 FG LUT.
__device__ __forceinline__ void sample_fg(const float* __restrict__ lut,
                                          float u, float v, float& f0, float& f1) {
    float fu = u * 256.0f - 0.5f;
    float fv = v * 256.0f - 0.5f;
    float x0f = floorf(fu), y0f = floorf(fv);
    float wx = fu - x0f, wy = fv - y0f;
    int x0 = iclamp((int)x0f, 0, 255);
    int y0 = iclamp((int)y0f, 0, 255);
    int x1 = iclamp(x0 + 1, 0, 255);
    int y1 = iclamp(y0 + 1, 0, 255);
    const float* p00 = lut + ((size_t)y0 * 256 + x0) * 2;
    const float* p01 = lut + ((size_t)y0 * 256 + x1) * 2;
    const float* p10 = lut + ((size_t)y1 * 256 + x0) * 2;
    const float* p11 = lut + ((size_t)y1 * 256 + x1) * 2;
    f0 = (p00[0]*(1.0f-wx) + p01[0]*wx)*(1.0f-wy) + (p10[0]*(1.0f-wx) + p11[0]*wx)*wy;
    f1 = (p00[1]*(1.0f-wx) + p01[1]*wx)*(1.0f-wy) + (p10[1]*(1.0f-wx) + p11[1]*wx)*wy;
}

struct SpecTab { const float* ptr[6]; };

#define DIFF_FLOATS (6 * 16 * 16 * 3)   // 4608 floats = 18432 bytes

// Full per-pixel shade given already-loaded G-buffer values.
__device__ __forceinline__ F3 shade_pixel(F3 pos, F3 nrm, F3 kdv, F3 ksv, F3 view,
                                          const float* s_diffuse,
                                          const unsigned long long* s_spec,
                                          const float* __restrict__ fg_lut) {
    F3 wo = normalize3(view - pos);
    float rough    = ksv.y;
    float metallic = ksv.z;
    F3 spec_col = kdv * metallic + (1.0f - metallic) * 0.04f;
    F3 diff_col = kdv * (1.0f - metallic);

    float wn = dot3(wo, nrm);
    F3 refl = normalize3(nrm * (2.0f * wn) - wo);

    // Diffuse IBL from the LDS-staged cubemap (ds_load path)
    F3 shaded = sample_cube(s_diffuse, 16, nrm) * diff_col;

    float NdotV = fmaxf(wn, 0.0001f);
    float f0, f1;
    sample_fg(fg_lut, NdotV, rough, f0, f1);

    // get_mip, then only the <=2 levels with nonzero tent weight.
    float lo  = (fminf(fmaxf(rough, MIN_R), MAX_R) - MIN_R) * (4.0f / (MAX_R - MIN_R));
    float hi  = (fminf(fmaxf(rough, MAX_R), 1.0f) - MAX_R) * (1.0f / (1.0f - MAX_R)) + 4.0f;
    float mip = (rough < MAX_R) ? lo : hi;        // in [0, 5]
    int   l0  = iclamp((int)floorf(mip), 0, 5);
    float w1  = mip - (float)l0;
    float w0  = 1.0f - w1;

    // Spec mip pointers come from a tiny LDS table: one ds_load_b64 per tap
    // instead of a v_cndmask chain over 6 kernarg pointers.
    const float* cm0 = (const float*)(uintptr_t)s_spec[l0];
    F3 spec_s = sample_cube(cm0, 512 >> l0, refl) * w0;
    int l1 = l0 + 1;
    if (l1 <= 5 && w1 > 0.0f) {
        const float* cm1 = (const float*)(uintptr_t)s_spec[l1];
        spec_s = spec_s + sample_cube(cm1, 512 >> l1, refl) * w1;
    }

    F3 reflectance = spec_col * f0 + f1;
    shaded = shaded + spec_s * reflectance;
    return shaded * (1.0f - ksv.x);
}

__global__ __launch_bounds__(256) void ibl_shade_kernel(
    const float* __restrict__ gb_pos,
    const float* __restrict__ gb_normal,
    const float* __restrict__ kd,
    const float* __restrict__ ks,
    const float* __restrict__ view_pos,
    const float* __restrict__ diffuse,
    const float* __restrict__ fg_lut,
    SpecTab spec,
    float* __restrict__ out,
    int n_pix, int hw, int blocks_per_batch)  // blocks_per_batch = hw/1024 if hw%1024==0 else 0
{
    __shared__ float s_diffuse[DIFF_FLOATS];
    __shared__ unsigned long long s_spec[6];

    // ---- Stage the 18KB diffuse cubemap into LDS via gfx1250 async copy ----
    {
        uint32_t lds_base = (uint32_t)(uintptr_t)(&s_diffuse[0]);
        const uint32_t nchunks = (DIFF_FLOATS * 4u) / 16u;  // 1152 x b128
        for (uint32_t idx = threadIdx.x; idx < nchunks; idx += 256u) {
            uint32_t lds_addr = lds_base + idx * 16u;
            uint32_t voff = idx * 16u;
            asm volatile("global_load_async_to_lds_b128 %0, %1, %2"
                         :: "v"(lds_addr), "v"(voff), "s"(diffuse)
                         : "memory");
        }
        if (threadIdx.x < 6)
            s_spec[threadIdx.x] = (unsigned long long)(uintptr_t)spec.ptr[threadIdx.x];
        asm volatile("s_wait_asynccnt 0" ::: "memory");
        __syncthreads();
    }

    const int base = blockIdx.x * 1024;          // 4 pixels per thread
    const int p0   = base + threadIdx.x * 4;

    if (blocks_per_batch > 0 && p0 + 3 < n_pix) {
        // ---- Fast path: uniform batch per block, vectorized b128 NT I/O ----
        int b = blockIdx.x / blocks_per_batch;   // scalar (SALU) division, uniform
        F3 view{view_pos[3*b + 0], view_pos[3*b + 1], view_pos[3*b + 2]};

        const size_t qb = 3 * (size_t)base;      // 16B-aligned (base % 1024 == 0)
        const int t3 = threadIdx.x * 3;

        // Speculative stream prefetch ~48KB ahead (global_prefetch_b8).
        __builtin_prefetch(gb_pos    + qb + t3 * 4 + 12288, 0, 0);
        __builtin_prefetch(gb_normal + qb + t3 * 4 + 12288, 0, 0);
        __builtin_prefetch(kd        + qb + t3 * 4 + 12288, 0, 0);
        __builtin_prefetch(ks        + qb + t3 * 4 + 12288, 0, 0);

        const f4* P = (const f4*)(gb_pos    + qb);
        const f4* N = (const f4*)(gb_normal + qb);
        const f4* K = (const f4*)(kd        + qb);
        const f4* S = (const f4*)(ks        + qb);

        // 4 pixels x 3 components = 3 x b128 nontemporal loads per array.
        f4 pa = __builtin_nontemporal_load(P + t3);
        f4 pb = __builtin_nontemporal_load(P + t3 + 1);
        f4 pc = __builtin_nontemporal_load(P + t3 + 2);
        f4 na = __builtin_nontemporal_load(N + t3);
        f4 nb = __builtin_nontemporal_load(N + t3 + 1);
        f4 nc = __builtin_nontemporal_load(N + t3 + 2);
        f4 ka = __builtin_nontemporal_load(K + t3);
        f4 kb = __builtin_nontemporal_load(K + t3 + 1);
        f4 kc = __builtin_nontemporal_load(K + t3 + 2);
        f4 sa = __builtin_nontemporal_load(S + t3);
        f4 sb = __builtin_nontemporal_load(S + t3 + 1);
        f4 sc = __builtin_nontemporal_load(S + t3 + 2);

        F3 o0 = shade_pixel(F3{pa.x,pa.y,pa.z}, F3{na.x,na.y,na.z},
                            F3{ka.x,ka.y,ka.z}, F3{sa.x,sa.y,sa.z},
                            view, s_diffuse, s_spec, fg_lut);
        F3 o1 = shade_pixel(F3{pa.w,pb.x,pb.y}, F3{na.w,nb.x,nb.y},
                            F3{ka.w,kb.x,kb.y}, F3{sa.w,sb.x,sb.y},
                            view, s_diffuse, s_spec, fg_lut);
        F3 o2 = shade_pixel(F3{pb.z,pb.w,pc.x}, F3{nb.z,nb.w,nc.x},
                            F3{kb.z,kb.w,kc.x}, F3{sb.z,sb.w,sc.x},
                            view, s_diffuse, s_spec, fg_lut);
        F3 o3 = shade_pixel(F3{pc.y,pc.z,pc.w}, F3{nc.y,nc.z,nc.w},
                            F3{kc.y,kc.z,kc.w}, F3{sc.y,sc.z,sc.w},
                            view, s_diffuse, s_spec, fg_lut);

        f4* O = (f4*)(out + qb);
        __builtin_nontemporal_store(f4{o0.x, o0.y, o0.z, o1.x}, O + t3);
        __builtin_nontemporal_store(f4{o1.y, o1.z, o2.x, o2.y}, O + t3 + 1);
        __builtin_nontemporal_store(f4{o2.z, o3.x, o3.y, o3.z}, O + t3 + 2);
    } else {
        // ---- Robust scalar tail path ----
        for (int j = 0; j < 4; ++j) {
            int p = p0 + j;
            if (p >= n_pix) break;
            size_t q = 3 * (size_t)p;
            int b = p / hw;
            F3 view{view_pos[3*b + 0], view_pos[3*b + 1], view_pos[3*b + 2]};
            F3 pos{gb_pos[q], gb_pos[q+1], gb_pos[q+2]};
            F3 nrm{gb_normal[q], gb_normal[q+1], gb_normal[q+2]};
            F3 kdv{kd[q], kd[q+1], kd[q+2]};
            F3 ksv{ks[q], ks[q+1], ks[q+2]};
            F3 o = shade_pixel(pos, nrm, kdv, ksv, view, s_diffuse, s_spec, fg_lut);
            out[q] = o.x; out[q+1] = o.y; out[q+2] = o.z;
        }
    }
}

extern "C" void kernel_launch(void* const* d_in, const int* in_sizes, int n_in,
                              void* d_out, int out_size, void* d_ws, size_t ws_size,
                              hipStream_t stream) {
    // setup_inputs() dict order:
    // 0 gb_pos, 1 gb_normal, 2 kd, 3 ks, 4 view_pos, 5 diffuse, 6 fg_lut,
    // 7..12 spec0..spec5
    const float* gb_pos    = (const float*)d_in[0];
    const float* gb_normal = (const float*)d_in[1];
    const float* kd        = (const float*)d_in[2];
    const float* ks        = (const float*)d_in[3];
    const float* view_pos  = (const float*)d_in[4];
    const float* diffuse   = (const float*)d_in[5];
    const float* fg_lut    = (const float*)d_in[6];

    SpecTab st;
    for (int i = 0; i < 6; ++i) st.ptr[i] = (const float*)d_in[7 + i];

    int n_pix = in_sizes[0] / 3;             // B*H*W
    int B     = in_sizes[4] / 3;             // view_pos is (B,1,1,3)
    int hw    = n_pix / (B > 0 ? B : 1);
    int bpb   = (hw > 0 && hw % 1024 == 0) ? hw / 1024 : 0;

    dim3 block(256);
    dim3 grid((n_pix + 1023) / 1024);        // 4 pixels per thread
    ibl_shade_kernel<<<grid, block, 0, stream>>>(
        gb_pos, gb_normal, kd, ks, view_pos, diffuse, fg_lut, st,
        (float*)d_out, n_pix, hw, bpb);
}